// ContextEncoding_31095563223520
// MI455X (gfx1250) — compile-verified
//
#include <hip/hip_runtime.h>

typedef __attribute__((ext_vector_type(2))) float v2f;
typedef __attribute__((ext_vector_type(4))) float v4f;
typedef __attribute__((ext_vector_type(8))) float v8f;

#define NBATCH 512
#define P_LEN  200
#define Q_LEN  60
#define Q_END  260
#define D_DIM  512

// LDS layout (floats). Strides padded for bank-conflict-free WMMA operand reads.
#define QS_ROWS    64
#define QS_STRIDE  532             // 512 + 20 (20 mod 64): conflict-free as B^T and B
#define SIM_ROWS   208
#define SIM_STRIDE 68              // 64 + 4
#define CH_ROWS    32              // ctx chunk rows (double-buffered)
#define CS_STRIDE  516             // 512 + 4
#define CS_BUF_FLOATS (CH_ROWS * CS_STRIDE)   // 16512 floats per buffer
#define NCHUNKS    7               // ceil(208/32); p-tiles 0..12

#define QS_OFF   0
#define SIM_OFF  (QS_ROWS * QS_STRIDE)                 // 34048
#define CS_OFF   (SIM_OFF + SIM_ROWS * SIM_STRIDE)     // 48192
#define LDS_FLOATS (CS_OFF + 2 * CS_BUF_FLOATS)        // 81216 floats = 324864 B

// phase>=2 scratch reuses the ctx-chunk region
#define MAXSIM_OFF CS_OFF
#define RED_OFF    (CS_OFF + 256)
#define Q2C_OFF    (CS_OFF + 512)

extern __shared__ float smem[];

// Low 32 bits of a generic LDS pointer == wave-relative LDS byte offset
__device__ __forceinline__ unsigned lds_addr32(const float* p) {
    return (unsigned)(unsigned long long)p;
}
// CDNA5 async DMA: global -> LDS, 16B per lane, tracked by ASYNCcnt (no VGPR round-trip)
__device__ __forceinline__ void async_copy_b128(float* lds_dst, const float* gsrc) {
    asm volatile("global_load_async_to_lds_b128 %0, %1, off"
                 :: "v"(lds_addr32(lds_dst)), "v"((unsigned long long)gsrc)
                 : "memory");
}
__device__ __forceinline__ void wait_async_all() {
    asm volatile("s_wait_asynccnt 0x0" ::: "memory");
}

__device__ __forceinline__ void stage_ctx_chunk(const float* __restrict__ ctx,
                                                float* buf, int chunk, int tid) {
    for (int i = tid; i < CH_ROWS * (D_DIM / 4); i += 256) {
        int r  = i >> 7;            // / 128
        int c4 = (i & 127) << 2;
        int p  = chunk * CH_ROWS + r;
        float* dst = buf + r * CS_STRIDE + c4;
        if (p < P_LEN) async_copy_b128(dst, ctx + p * D_DIM + c4);
        else           *(v4f*)dst = (v4f){0.f, 0.f, 0.f, 0.f};
    }
}

__global__ __launch_bounds__(256, 1)
void bidaf_attention_kernel(const float* __restrict__ enc, float* __restrict__ out) {
    const int b    = blockIdx.x;
    const int tid  = threadIdx.x;
    const int lane = tid & 31;
    const int wave = tid >> 5;
    const int m16  = lane & 15;     // row (A) / col (B,C,D) within 16x16 tile
    const int hi   = lane >> 4;     // half-wave select
    const int koff = hi << 1;       // k-pair offset for f32 WMMA A/B operands

    const float* __restrict__ ctx = enc + (size_t)b * (Q_END * D_DIM); // rows 0..199
    const float* __restrict__ qry = ctx + (size_t)P_LEN * D_DIM;       // rows 200..259

    float* qs     = smem + QS_OFF;
    float* simS   = smem + SIM_OFF;
    float* csBuf0 = smem + CS_OFF;
    float* csBuf1 = smem + CS_OFF + CS_BUF_FLOATS;
    float* maxsim = smem + MAXSIM_OFF;
    float* red    = smem + RED_OFF;
    float* q2c    = smem + Q2C_OFF;

    // ---- Phase 0: async-stage qry (60x512) + ctx chunk 0; zero LDS pad rows ----
    for (int i = tid; i < QS_ROWS * (D_DIM / 4); i += 256) {
        int r  = i >> 7;
        int c4 = (i & 127) << 2;
        float* dst = qs + r * QS_STRIDE + c4;
        if (r < Q_LEN) async_copy_b128(dst, qry + r * D_DIM + c4);
        else           *(v4f*)dst = (v4f){0.f, 0.f, 0.f, 0.f};
    }
    stage_ctx_chunk(ctx, csBuf0, 0, tid);
    wait_async_all();
    __syncthreads();

    // ---- Phase 1: sim = ctx @ qry^T (200x60), double-buffered ctx chunks ----
    for (int chunk = 0; chunk < NCHUNKS; ++chunk) {
        float* cur = (chunk & 1) ? csBuf1 : csBuf0;
        if (chunk + 1 < NCHUNKS)   // prefetch next chunk while computing this one
            stage_ctx_chunk(ctx, ((chunk + 1) & 1) ? csBuf1 : csBuf0, chunk + 1, tid);

        // p-tiles in this chunk: 2 (or 1 for the last chunk -> p-tile 12 only)
        const int njobs = (chunk < NCHUNKS - 1) ? 8 : 4;   // ptiles*4 qtiles (uniform)
        for (int job = wave; job < njobs; job += 8) {
            const int ptl = job >> 2;     // local p-tile 0..1
            const int qt  = job & 3;
            v8f acc = {};
            const float* aBase = cur + (ptl * 16 + m16) * CS_STRIDE; // ctx row m
            const float* bBase = qs  + (qt  * 16 + m16) * QS_STRIDE; // qry row n (B = qry^T)
            #pragma unroll 8
            for (int k0 = 0; k0 < D_DIM; k0 += 4) {
                v2f a  = *(const v2f*)(aBase + k0 + koff);
                v2f bb = *(const v2f*)(bBase + k0 + koff);
                acc = __builtin_amdgcn_wmma_f32_16x16x4_f32(
                        false, a, false, bb, (short)0, acc, false, false);
            }
            float* sRow = simS + (chunk * CH_ROWS + ptl * 16) * SIM_STRIDE + qt * 16 + m16;
            #pragma unroll
            for (int v = 0; v < 8; ++v)
                sRow[(v + 8 * hi) * SIM_STRIDE] = acc[v];
        }
        wait_async_all();
        __syncthreads();
    }

    // ---- Phase 2: per-row softmax over q (c2q_w) + row max (maxsim) ----
    for (int p = tid; p < SIM_ROWS; p += 256) {
        float* row = simS + p * SIM_STRIDE;
        if (p < P_LEN) {
            float mx = row[0];
            for (int q = 1; q < Q_LEN; ++q) mx = fmaxf(mx, row[q]);
            maxsim[p] = mx;
            float s = 0.f;
            for (int q = 0; q < Q_LEN; ++q) { float e = __expf(row[q] - mx); row[q] = e; s += e; }
            float inv = 1.f / s;
            for (int q = 0;     q < Q_LEN; ++q) row[q] *= inv;
            for (int q = Q_LEN; q < 64;    ++q) row[q]  = 0.f;
        } else {
            for (int q = 0; q < 64; ++q) row[q] = 0.f;
        }
    }
    __syncthreads();

    // ---- Phase 3: softmax over p of maxsim -> q2c weights ----
    red[tid] = (tid < P_LEN) ? maxsim[tid] : -3.4e38f;
    __syncthreads();
    for (int s = 128; s > 0; s >>= 1) {
        if (tid < s) red[tid] = fmaxf(red[tid], red[tid + s]);
        __syncthreads();
    }
    const float gmax = red[0];
    __syncthreads();
    const float e = (tid < P_LEN) ? __expf(maxsim[tid] - gmax) : 0.f;
    red[tid] = e;
    __syncthreads();
    for (int s = 128; s > 0; s >>= 1) {
        if (tid < s) red[tid] += red[tid + s];
        __syncthreads();
    }
    const float ginv = 1.f / red[0];
    __syncthreads();
    if (tid < P_LEN) maxsim[tid] = e * ginv;
    __syncthreads();

    // ---- Phase 4: q2c_vec[d] = sum_p w[p] * ctx[p][d]  (coalesced, L2-hit) ----
    {
        float s0 = 0.f, s1 = 0.f;
        for (int p = 0; p < P_LEN; ++p) {
            const float w = maxsim[p];
            s0 += w * ctx[p * D_DIM + tid];
            s1 += w * ctx[p * D_DIM + tid + 256];
        }
        q2c[tid]       = s0;
        q2c[tid + 256] = s1;
    }
    __syncthreads();

    // ---- Phase 5: c2q = c2q_w @ qry, fused 4-section output (NT stores) ----
    const size_t obase = (size_t)b * P_LEN * 2048;
    for (int job = wave; job < 13 * 32; job += 8) {   // 416 tiles, 52/wave (uniform)
        const int pt = job >> 5;
        const int dt = job & 31;
        const int p0 = pt * 16, d0 = dt * 16;
        v8f acc = {};
        const float* aBase = simS + (p0 + m16) * SIM_STRIDE;  // A = c2q_w rows
        const float* bCol  = qs + d0 + m16;                   // B = qry (row-major)
        #pragma unroll
        for (int k0 = 0; k0 < 64; k0 += 4) {
            v2f a = *(const v2f*)(aBase + k0 + koff);
            v2f bb;
            bb.x = bCol[(k0 + koff)     * QS_STRIDE];
            bb.y = bCol[(k0 + koff + 1) * QS_STRIDE];
            acc = __builtin_amdgcn_wmma_f32_16x16x4_f32(
                    false, a, false, bb, (short)0, acc, false, false);
        }
        const int d = d0 + m16;
        const float q2cv = q2c[d];
        #pragma unroll
        for (int v = 0; v < 8; ++v) {
            const int p = p0 + v + 8 * hi;
            if (p < P_LEN) {
                const float c2qv = acc[v];
                const float ctxv = ctx[p * D_DIM + d];
                float* o = out + obase + (size_t)p * 2048 + d;
                __builtin_nontemporal_store(ctxv,        o);          // ctx
                __builtin_nontemporal_store(c2qv,        o + 512);    // c2q
                __builtin_nontemporal_store(ctxv * c2qv, o + 1024);   // ctx * c2q
                __builtin_nontemporal_store(ctxv * q2cv, o + 1536);   // ctx * q2c
            }
        }
    }
}

extern "C" void kernel_launch(void* const* d_in, const int* in_sizes, int n_in,
                              void* d_out, int out_size, void* d_ws, size_t ws_size,
                              hipStream_t stream) {
    const float* enc = (const float*)d_in[0];
    float* out = (float*)d_out;
    const size_t smem_bytes = (size_t)LDS_FLOATS * sizeof(float); // 324,864 B
    hipFuncSetAttribute((const void*)bidaf_attention_kernel,
                        hipFuncAttributeMaxDynamicSharedMemorySize, (int)smem_bytes);
    bidaf_attention_kernel<<<NBATCH, 256, smem_bytes, stream>>>(enc, out);
}